// SpatialEncoding_33191507263825
// MI455X (gfx1250) — compile-verified
//
#include <hip/hip_runtime.h>

typedef float vfloat4 __attribute__((ext_vector_type(4)));
typedef int   v4i     __attribute__((ext_vector_type(4)));

// CDNA5 async LDS->global store path (gfx1250 only; probe via __has_builtin so
// host pass and older toolchains fall back to plain b128 stores).
#if defined(__HIP_DEVICE_COMPILE__) && \
    __has_builtin(__builtin_amdgcn_global_store_async_from_lds_b128) && \
    __has_builtin(__builtin_amdgcn_s_wait_asynccnt)
#define ASYNC_ZERO 1
// Pointer-to-AS1-v4i and pointer-to-AS3-v4i, matching the builtin's prototype.
typedef __attribute__((address_space(1))) v4i* gv4i_ptr;
typedef __attribute__((address_space(3))) v4i* lv4i_ptr;
#else
#define ASYNC_ZERO 0
#endif

// ---------------------------------------------------------------------------
// Kernel 1: zero the 256 MB output at HBM line rate.
// Async stores stream 16B/lane straight out of a zeroed LDS staging buffer:
// no VGPR data operands, tracked on ASYNCcnt, drained before s_endpgm.
// ---------------------------------------------------------------------------
__global__ __launch_bounds__(256)
void spatial_zero_kernel(vfloat4* __restrict__ out4, long long n4,
                         float* __restrict__ out, long long n) {
  const long long tid    = (long long)blockIdx.x * blockDim.x + threadIdx.x;
  const long long stride = (long long)gridDim.x * blockDim.x;

#if ASYNC_ZERO
  __shared__ v4i stage[256];
  v4i zi = {0, 0, 0, 0};  // bit pattern of 0.0f
  stage[threadIdx.x] = zi;
  __syncthreads();  // LDS zeros visible before async stores read them
  for (long long i = tid; i < n4; i += stride) {
    __builtin_amdgcn_global_store_async_from_lds_b128(
        (gv4i_ptr)(out4 + i), (lv4i_ptr)&stage[threadIdx.x],
        /*offset=*/0, /*cpol=*/0);
  }
  __builtin_amdgcn_s_wait_asynccnt(0);
#else
  const vfloat4 z = {0.f, 0.f, 0.f, 0.f};
  for (long long i = tid; i < n4; i += stride) {
    out4[i] = z;  // global_store_b128
  }
#endif

  // Scalar tail (out_size not divisible by 4 — not expected here, but safe).
  for (long long i = n4 * 4 + tid; i < n; i += stride) {
    out[i] = 0.0f;
  }
}

// ---------------------------------------------------------------------------
// Kernel 2: scatter b[clip(path_len,1,5)-1] into out[src*N + dst].
// 4 edges per thread via b128 index loads; b[] hoisted to SGPRs and selected
// with a v_cndmask chain; duplicate (src,dst) pairs are undefined-order in the
// reference, so plain last-write-wins stores are the faithful semantics.
// ---------------------------------------------------------------------------
__global__ __launch_bounds__(256)
void spatial_scatter_kernel(const int* __restrict__ src,
                            const int* __restrict__ dst,
                            const int* __restrict__ plen,
                            const float* __restrict__ b,
                            float* __restrict__ out, int E, int N) {
  // Tiny uniform table -> scalar loads, lives in SGPRs.
  const float b0 = b[0], b1 = b[1], b2 = b[2], b3 = b[3], b4 = b[4];

  const int i = (blockIdx.x * blockDim.x + threadIdx.x) * 4;
  if (i >= E) return;

  // Stream-ahead prefetch of the three index arrays (global_prefetch_b8;
  // speculative TH -> OOB addresses past the end are silently dropped).
  __builtin_prefetch(src  + i + 8192, 0, 0);
  __builtin_prefetch(dst  + i + 8192, 0, 0);
  __builtin_prefetch(plen + i + 8192, 0, 0);

  auto pick = [&](int p) -> float {
    int idx = p < 1 ? 0 : (p > 5 ? 4 : p - 1);  // clip(p,1,5)-1
    float v = b0;
    v = (idx == 1) ? b1 : v;
    v = (idx == 2) ? b2 : v;
    v = (idx == 3) ? b3 : v;
    v = (idx == 4) ? b4 : v;
    return v;
  };

  if (i + 3 < E) {
    const int4 s4 = *(const int4*)(src  + i);
    const int4 d4 = *(const int4*)(dst  + i);
    const int4 p4 = *(const int4*)(plen + i);
    out[(size_t)s4.x * N + d4.x] = pick(p4.x);
    out[(size_t)s4.y * N + d4.y] = pick(p4.y);
    out[(size_t)s4.z * N + d4.z] = pick(p4.z);
    out[(size_t)s4.w * N + d4.w] = pick(p4.w);
  } else {
    for (int k = i; k < E; ++k) {
      out[(size_t)src[k] * N + dst[k]] = pick(plen[k]);
    }
  }
}

// ---------------------------------------------------------------------------
// Host entry point.
// Inputs (setup_inputs order): x [N,128] f32 (unused), b [5] f32,
// src [E] int, dst [E] int, path_len [E] int. Output: [N,N] f32.
// ---------------------------------------------------------------------------
extern "C" void kernel_launch(void* const* d_in, const int* in_sizes, int n_in,
                              void* d_out, int out_size, void* d_ws, size_t ws_size,
                              hipStream_t stream) {
  (void)n_in; (void)d_ws; (void)ws_size;

  const float* b    = (const float*)d_in[1];
  const int*   src  = (const int*)d_in[2];
  const int*   dst  = (const int*)d_in[3];
  const int*   plen = (const int*)d_in[4];
  float*       out  = (float*)d_out;

  const int E = in_sizes[2];
  const int N = in_sizes[0] / 128;  // x is [N,128]; only its shape is used

  const long long n  = (long long)out_size;
  const long long n4 = n / 4;

  // Zero pass: 4M threads grid-striding over 16M float4 slots (4 iters each).
  spatial_zero_kernel<<<16384, 256, 0, stream>>>((vfloat4*)out, n4, out, n);

  // Scatter pass: one thread per 4 edges.
  const long long quads  = ((long long)E + 3) / 4;
  const int       blocks = (int)((quads + 255) / 256);
  spatial_scatter_kernel<<<blocks, 256, 0, stream>>>(src, dst, plen, b, out, E, N);
}